// Heatmap_Ball_Detection_Loss_Weighted_MultiFrame_26096221290542
// MI455X (gfx1250) — compile-verified
//
#include <hip/hip_runtime.h>

typedef __attribute__((ext_vector_type(2))) float v2f;
typedef __attribute__((ext_vector_type(8))) float v8f;

#define SIGMA_F   0.3f
#define EPS_NORM  1e-8f
#define LN2F      0.69314718056f

// Raw hardware log2 (single v_log_f32 TRANS op; no denorm fixup code).
// Valid here because 1-p is always in (1e-4, 1): no zero/denorm/inf inputs.
__device__ __forceinline__ float raw_log2(float x) {
    return __builtin_amdgcn_logf(x);
}

// ---------------------------------------------------------------------------
// Kernel 1: one block per frame (b,n). Streams the full x-row (W floats) and
// y-row (H floats) computing sum of -log(1-p)  (the BCE-against-zero base
// term), reduces across the block, writes one float per frame to d_ws.
// Memory-bound: 49 MB total at 23.3 TB/s ~= 2.1 us floor; per-float4 work is
// 4x v_sub + 4x v_log_f32 (TRANS, co-executes) + 4x v_add.
// ---------------------------------------------------------------------------
__global__ void hm_frame_base_kernel(const float* __restrict__ pred_x,
                                     const float* __restrict__ pred_y,
                                     float* __restrict__ frame_sum,
                                     int W, int H) {
    const int frame = blockIdx.x;
    const int tid   = threadIdx.x;
    const int nthr  = blockDim.x;

    const float* rx = pred_x + (size_t)frame * (size_t)W;
    const float* ry = pred_y + (size_t)frame * (size_t)H;

    float acc = 0.0f;   // accumulates sum of log2(1-p); scaled by -ln2 at end

    // ---- x row (b128 loads + prefetch) ----
    const int W4 = W >> 2;
    const float4* rx4 = (const float4*)rx;
    for (int i = tid; i < W4; i += nthr) {
        __builtin_prefetch(rx4 + i + nthr, 0, 1);   // global_prefetch_b8
        float4 v = rx4[i];
        float l0 = raw_log2(1.0f - v.x);
        float l1 = raw_log2(1.0f - v.y);
        float l2 = raw_log2(1.0f - v.z);
        float l3 = raw_log2(1.0f - v.w);
        acc += (l0 + l1) + (l2 + l3);
    }
    for (int i = (W4 << 2) + tid; i < W; i += nthr)
        acc += raw_log2(1.0f - rx[i]);

    // ---- y row ----
    const int H4 = H >> 2;
    const float4* ry4 = (const float4*)ry;
    for (int i = tid; i < H4; i += nthr) {
        __builtin_prefetch(ry4 + i + nthr, 0, 1);
        float4 v = ry4[i];
        float l0 = raw_log2(1.0f - v.x);
        float l1 = raw_log2(1.0f - v.y);
        float l2 = raw_log2(1.0f - v.z);
        float l3 = raw_log2(1.0f - v.w);
        acc += (l0 + l1) + (l2 + l3);
    }
    for (int i = (H4 << 2) + tid; i < H; i += nthr)
        acc += raw_log2(1.0f - ry[i]);

    // convert sum(log2(1-p)) -> sum(-ln(1-p))
    acc *= -LN2F;

    // ---- wave32 reduction via one fp32 WMMA against a ones matrix ----
    // A (16x4 f32 layout): lanes 0-15 hold {K0,K1}, lanes 16-31 hold {K2,K3}.
    // With a.y = 0 every lane's partial occupies exactly one A slot.
    // B = ones(4x16)  =>  D[m][n] = rowsum(A_m) for every n.
    // Per-lane sum of the 8 D VGPRs = sum(rows 0-7) on lanes 0-15,
    // sum(rows 8-15) on lanes 16-31; shfl_xor(16) completes the wave total.
    v2f a; a.x = acc;  a.y = 0.0f;
    v2f b; b.x = 1.0f; b.y = 1.0f;
    v8f c = {};
    c = __builtin_amdgcn_wmma_f32_16x16x4_f32(false, a, false, b,
                                              (short)0, c, false, false);
    float s = c[0] + c[1] + c[2] + c[3] + c[4] + c[5] + c[6] + c[7];
    s += __shfl_xor(s, 16, 32);

    __shared__ float wavesum[32];
    const int lane = tid & 31;
    const int wave = tid >> 5;
    if (lane == 0) wavesum[wave] = s;
    __syncthreads();
    if (tid == 0) {
        float tot = 0.0f;
        const int nw = nthr >> 5;
        for (int i = 0; i < nw; ++i) tot += wavesum[i];
        frame_sum[frame] = tot;
    }
}

// ---------------------------------------------------------------------------
// Kernel 2 (single block): sparse per-frame corrections + masking + weights +
// final normalization.  BCE(p,t) - BCE(p,0) = -t * (log p - log(1-p)), and the
// sigma=0.3 Gaussian is numerically zero beyond |d|=4, so each frame touches
// <= 9 positions per axis.  Deterministic block-tree reduction, no atomics.
// ---------------------------------------------------------------------------
__global__ void hm_finalize_kernel(const float* __restrict__ pred_x,
                                   const float* __restrict__ pred_y,
                                   const int*   __restrict__ target,  // [BN,2]
                                   const int*   __restrict__ vis,     // [BN]
                                   const float* __restrict__ frame_sum,
                                   float* __restrict__ out,
                                   int BN, int W, int H) {
    __shared__ float s_tot[256];
    __shared__ int   s_cnt[256];
    const int tid = threadIdx.x;

    const float weights[4] = {1.0f, 2.0f, 2.0f, 3.0f};
    const float inv2s2 = 0.5f / (SIGMA_F * SIGMA_F);
    const int   R = 4;   // exp(-5.5556*d^2) == 0 in f32 beyond |d| = 4

    float total = 0.0f;
    int   cnt   = 0;

    for (int f = tid; f < BN; f += (int)blockDim.x) {
        const int rtx = target[2 * f];
        const int rty = target[2 * f + 1];
        const bool skip = (rtx == 0) && (rty == 0);
        if (skip) continue;

        cnt++;
        const int  v     = vis[f];
        const bool gauss = (v == 3);

        float loss = frame_sum[f];

        // two axes: (pred_x row, rtx, W) and (pred_y row, rty, H)
        for (int axis = 0; axis < 2; ++axis) {
            const int    S   = (axis == 0) ? W : H;
            const int    raw = (axis == 0) ? rtx : rty;
            const float* row = (axis == 0) ? pred_x + (size_t)f * (size_t)W
                                           : pred_y + (size_t)f * (size_t)H;
            const int t = min(max(raw, 0), S - 1);
            if (gauss) {
                float G = 0.0f;
                for (int d = -R; d <= R; ++d) {
                    const int w = t + d;
                    if (w < 0 || w >= S) continue;
                    G += __expf(-inv2s2 * (float)(d * d));
                }
                const float inv = 1.0f / (G + EPS_NORM);
                for (int d = -R; d <= R; ++d) {
                    const int w = t + d;
                    if (w < 0 || w >= S) continue;
                    const float tv = __expf(-inv2s2 * (float)(d * d)) * inv;
                    const float p  = row[w];
                    loss -= tv * (__logf(p) - __logf(1.0f - p));
                }
            } else {
                const float p = row[t];
                loss -= (__logf(p) - __logf(1.0f - p));
            }
        }

        total += weights[v] * loss;
    }

    s_tot[tid] = total;
    s_cnt[tid] = cnt;
    __syncthreads();
    for (int off = 128; off > 0; off >>= 1) {
        if (tid < off) {
            s_tot[tid] += s_tot[tid + off];
            s_cnt[tid] += s_cnt[tid + off];
        }
        __syncthreads();
    }
    if (tid == 0) {
        out[0] = (s_cnt[0] == 0) ? 0.0f : (s_tot[0] / (float)s_cnt[0]);
    }
}

// ---------------------------------------------------------------------------
extern "C" void kernel_launch(void* const* d_in, const int* in_sizes, int n_in,
                              void* d_out, int out_size, void* d_ws, size_t ws_size,
                              hipStream_t stream) {
    const float* pred_x = (const float*)d_in[0];   // [B,N,W] f32
    const float* pred_y = (const float*)d_in[1];   // [B,N,H] f32
    const int*   target = (const int*)d_in[2];     // [B,N,2] i32
    const int*   vis    = (const int*)d_in[3];     // [B,N]   i32

    const int BN = in_sizes[3];            // B*N (= 4096)
    const int W  = in_sizes[0] / BN;       // 1920
    const int H  = in_sizes[1] / BN;       // 1080

    float* frame_sum = (float*)d_ws;       // BN floats of scratch

    hm_frame_base_kernel<<<BN, 256, 0, stream>>>(pred_x, pred_y, frame_sum, W, H);
    hm_finalize_kernel<<<1, 256, 0, stream>>>(pred_x, pred_y, target, vis,
                                              frame_sum, (float*)d_out, BN, W, H);
}